// GCN_12025908429355
// MI455X (gfx1250) — compile-verified
//
#include <hip/hip_runtime.h>
#include <hip/hip_bf16.h>
#include <math.h>

typedef __attribute__((ext_vector_type(2))) float v2f;
typedef __attribute__((ext_vector_type(8))) float v8f;

#define D_IN   128
#define D_H    256
#define D_OUT  16

// ---------------------------------------------------------------------------
// Degree / norm kernels
// ---------------------------------------------------------------------------
__global__ void k_deg_init(float* deg, int n) {
    int i = blockIdx.x * blockDim.x + threadIdx.x;
    if (i < n) deg[i] = 1.0f;               // self-loop contributes 1
}

__global__ void k_deg_count(const int* __restrict__ dst, float* deg, int E) {
    int i = blockIdx.x * blockDim.x + threadIdx.x;
    if (i < E) unsafeAtomicAdd(&deg[dst[i]], 1.0f);   // global_atomic_add_f32
}

__global__ void k_dinv(float* deg, int n) {
    int i = blockIdx.x * blockDim.x + threadIdx.x;
    if (i < n) deg[i] = rsqrtf(deg[i]);     // deg >= 1 always (self-loop)
}

// ---------------------------------------------------------------------------
// GEMM1: h1[N,256] = x[N,128] @ W1[128,256]   via V_WMMA_F32_16X16X4_F32
// 512 threads = 16 waves; wave w owns N-tile w; block owns one 16-row M-tile.
// ---------------------------------------------------------------------------
__global__ __launch_bounds__(512) void k_gemm1(const float* __restrict__ x,
                                               const float* __restrict__ W1,
                                               float* __restrict__ h1) {
    const int wave = threadIdx.x >> 5;      // 0..15  -> N tile
    const int lane = threadIdx.x & 31;
    const int half = lane >> 4;             // 0/1
    const int mr   = lane & 15;
    const int m0   = blockIdx.x * 16;
    const int n0   = wave * 16;

    const float* arow = x  + (size_t)(m0 + mr) * D_IN + 2 * half;   // A: row mr, K=2*half+{0,1}
    const float* bcol = W1 + (size_t)(2 * half) * D_H + (n0 + mr);  // B: col mr, K rows

    v8f c = {};
    #pragma unroll 4
    for (int kb = 0; kb < D_IN; kb += 4) {
        v2f a = *(const v2f*)(arow + kb);                  // b64 load, K-adjacent
        v2f b;
        b.x = bcol[(size_t)kb * D_H];
        b.y = bcol[(size_t)kb * D_H + D_H];
        c = __builtin_amdgcn_wmma_f32_16x16x4_f32(
                false, a, false, b, (short)0, c, false, false);
    }
    // C/D layout: VGPR r -> row m0 + r + 8*half, col n0 + mr
    float* out = h1 + (size_t)(m0 + 8 * half) * D_H + (n0 + mr);
    #pragma unroll
    for (int r = 0; r < 8; ++r) out[(size_t)r * D_H] = c[r];
}

// ---------------------------------------------------------------------------
// Aggregation layer 1: agg1 = dinv[i]^2 * h1[i]  (self-loop init, no atomics)
// then per-edge scatter-add of dinv[s]*dinv[d]*h1[s]
// ---------------------------------------------------------------------------
__global__ void k_agg1_self(const float* __restrict__ h1,
                            const float* __restrict__ dinv,
                            float* __restrict__ agg1, int n) {
    int idx = blockIdx.x * blockDim.x + threadIdx.x;
    if (idx < n * D_H) {
        float di = dinv[idx >> 8];          // D_H == 256
        agg1[idx] = di * di * h1[idx];
    }
}

__global__ void k_agg1_edges(const int* __restrict__ src,
                             const int* __restrict__ dst,
                             const float* __restrict__ dinv,
                             const float* __restrict__ h1,
                             float* __restrict__ agg1, int E) {
    int tid  = blockIdx.x * blockDim.x + threadIdx.x;
    int e    = tid >> 5;                    // one wave per edge
    int lane = tid & 31;
    if (e >= E) return;                     // wave-uniform
    int s = src[e], d = dst[e];
    float w = dinv[s] * dinv[d];
    const float* hs = h1   + (size_t)s * D_H;
    float*       ad = agg1 + (size_t)d * D_H;
    #pragma unroll
    for (int k = lane; k < D_H; k += 32)
        unsafeAtomicAdd(&ad[k], hs[k] * w);
}

// ---------------------------------------------------------------------------
// GEMM2: g2[N,16] = relu(agg1 + b1) @ W2[256,16]   (relu+bias fused in A load)
// 256 threads = 8 waves; wave owns one 16-row M-tile, single N-tile.
// ---------------------------------------------------------------------------
__global__ __launch_bounds__(256) void k_gemm2(const float* __restrict__ agg1,
                                               const float* __restrict__ b1,
                                               const float* __restrict__ W2,
                                               float* __restrict__ g2, int ntiles) {
    const int wave = threadIdx.x >> 5;
    const int lane = threadIdx.x & 31;
    const int tile = blockIdx.x * 8 + wave;
    if (tile >= ntiles) return;             // wave-uniform: EXEC all-1s for WMMA
    const int half = lane >> 4;
    const int mr   = lane & 15;
    const int m0   = tile * 16;

    const float* arow = agg1 + (size_t)(m0 + mr) * D_H + 2 * half;
    const float* brow = W2   + (size_t)(2 * half) * D_OUT + mr;     // n0 = 0

    v8f c = {};
    #pragma unroll 4
    for (int kb = 0; kb < D_H; kb += 4) {
        int k0 = kb + 2 * half;
        v2f a;
        a.x = fmaxf(arow[kb]     + b1[k0],     0.0f);   // fused relu(h + b1)
        a.y = fmaxf(arow[kb + 1] + b1[k0 + 1], 0.0f);
        v2f b;
        b.x = brow[(size_t)kb * D_OUT];
        b.y = brow[(size_t)kb * D_OUT + D_OUT];
        c = __builtin_amdgcn_wmma_f32_16x16x4_f32(
                false, a, false, b, (short)0, c, false, false);
    }
    float* out = g2 + (size_t)(m0 + 8 * half) * D_OUT + mr;
    #pragma unroll
    for (int r = 0; r < 8; ++r) out[(size_t)r * D_OUT] = c[r];
}

// ---------------------------------------------------------------------------
// Aggregation layer 2 (16-wide messages)
// ---------------------------------------------------------------------------
__global__ void k_agg2_self(const float* __restrict__ g2,
                            const float* __restrict__ dinv,
                            float* __restrict__ agg2, int n) {
    int idx = blockIdx.x * blockDim.x + threadIdx.x;
    if (idx < n * D_OUT) {
        float di = dinv[idx >> 4];
        agg2[idx] = di * di * g2[idx];
    }
}

__global__ void k_agg2_edges(const int* __restrict__ src,
                             const int* __restrict__ dst,
                             const float* __restrict__ dinv,
                             const float* __restrict__ g2,
                             float* __restrict__ agg2, int E) {
    int tid = blockIdx.x * blockDim.x + threadIdx.x;
    int e = tid >> 4;                       // 16 lanes per edge
    int k = tid & 15;
    if (e >= E) return;
    int s = src[e], d = dst[e];
    float w = dinv[s] * dinv[d];
    unsafeAtomicAdd(&agg2[(size_t)d * D_OUT + k], g2[(size_t)s * D_OUT + k] * w);
}

// ---------------------------------------------------------------------------
// log_softmax(agg2 + b2) over dim 16 — one thread per row
// ---------------------------------------------------------------------------
__global__ void k_logsoftmax(const float* __restrict__ agg2,
                             const float* __restrict__ b2,
                             float* __restrict__ out, int n) {
    int row = blockIdx.x * blockDim.x + threadIdx.x;
    if (row >= n) return;
    float v[D_OUT];
    float m = -INFINITY;
    #pragma unroll
    for (int k = 0; k < D_OUT; ++k) {
        v[k] = agg2[(size_t)row * D_OUT + k] + b2[k];
        m = fmaxf(m, v[k]);
    }
    float s = 0.0f;
    #pragma unroll
    for (int k = 0; k < D_OUT; ++k) s += __expf(v[k] - m);
    float lse = m + __logf(s);
    #pragma unroll
    for (int k = 0; k < D_OUT; ++k)
        out[(size_t)row * D_OUT + k] = v[k] - lse;
}

// ---------------------------------------------------------------------------
// Host launcher
// ---------------------------------------------------------------------------
extern "C" void kernel_launch(void* const* d_in, const int* in_sizes, int n_in,
                              void* d_out, int out_size, void* d_ws, size_t ws_size,
                              hipStream_t stream) {
    const float* x    = (const float*)d_in[0];
    const int*   ei   = (const int*)d_in[1];     // [2,E]: row0 = src, row1 = dst
    const float* W1   = (const float*)d_in[2];
    const float* b1   = (const float*)d_in[3];
    const float* W2   = (const float*)d_in[4];
    const float* b2   = (const float*)d_in[5];
    float*       outp = (float*)d_out;

    const int N = in_sizes[0] / D_IN;            // 50000
    const int E = in_sizes[1] / 2;               // 800000
    const int* src = ei;
    const int* dst = ei + E;

    // workspace layout (floats)
    float* ws   = (float*)d_ws;
    float* dinv = ws;                              // N
    float* h1   = dinv + N;                        // N*256
    float* agg1 = h1   + (size_t)N * D_H;          // N*256
    float* g2   = agg1 + (size_t)N * D_H;          // N*16
    float* agg2 = g2   + (size_t)N * D_OUT;        // N*16

    const int ntiles = N / 16;                     // 3125 (N % 16 == 0)

    // 1) degrees -> dinv
    k_deg_init <<<(N + 255) / 256, 256, 0, stream>>>(dinv, N);
    k_deg_count<<<(E + 255) / 256, 256, 0, stream>>>(dst, dinv, E);
    k_dinv     <<<(N + 255) / 256, 256, 0, stream>>>(dinv, N);

    // 2) h1 = x @ W1  (WMMA f32 16x16x4)
    k_gemm1<<<ntiles, 512, 0, stream>>>(x, W1, h1);

    // 3) symmetric-norm aggregation over h1
    k_agg1_self <<<(N * D_H + 255) / 256, 256, 0, stream>>>(h1, dinv, agg1, N);
    k_agg1_edges<<<(E * 32 + 255) / 256, 256, 0, stream>>>(src, dst, dinv, h1, agg1, E);

    // 4) g2 = relu(agg1 + b1) @ W2  (WMMA f32 16x16x4, fused relu+bias)
    k_gemm2<<<(ntiles + 7) / 8, 256, 0, stream>>>(agg1, b1, W2, g2, ntiles);

    // 5) aggregation over g2
    k_agg2_self <<<(N * D_OUT + 255) / 256, 256, 0, stream>>>(g2, dinv, agg2, N);
    k_agg2_edges<<<(E * 16 + 255) / 256, 256, 0, stream>>>(src, dst, dinv, g2, agg2, E);

    // 6) log_softmax(agg2 + b2) -> out
    k_logsoftmax<<<(N + 255) / 256, 256, 0, stream>>>(agg2, b2, outp, N);
}